// NDP_27882927685986
// MI455X (gfx1250) — compile-verified
//
#include <hip/hip_runtime.h>
#include <hip/hip_bf16.h>
#include <math.h>

// ---------------------------------------------------------------------------
// Problem constants: N=1024 nodes, D=H=256.
// Outputs (flat, concatenated): u (1024*256) | repl (1024) | edge (1024*1024)
// ---------------------------------------------------------------------------

typedef __attribute__((ext_vector_type(2))) float v2f;
typedef __attribute__((ext_vector_type(8))) float v8f;

#define GN 1024
#define GD 256

// ---------------------------------------------------------------------------
// WMMA fp32 GEMM:  Y[m][n] = act( sum_k X[m][k] * W[n][k] + bias[n] )
//   X: M x K row-major (row stride K)
//   W: N x K row-major with row stride ldw (supports Wu1 column slices)
//   Uses V_WMMA_F32_16X16X4_F32: one wave owns one 16x16 output tile.
//   fp32 A/B/C keeps full reference precision.
//
// VGPR layouts (ISA 7.12.2, wave32):
//   A 16x4:  lane l -> m = l&15, half = l>>4; v2f covers k0+2*half .. +1
//   B 4x16:  lane l -> n = l&15, half = l>>4; v2f covers k0+2*half .. +1
//   C 16x16: vgpr r -> m = r + 8*half, n = l&15
// ---------------------------------------------------------------------------
__global__ void wmma_gemm_f32(const float* __restrict__ X,
                              const float* __restrict__ W,
                              const float* __restrict__ bias,
                              float* __restrict__ Y,
                              int M, int Ncols, int K, int ldw, int relu)
{
    const int lane   = threadIdx.x & 31;
    const int waveId = threadIdx.x >> 5;
    const int tilesN = Ncols >> 4;

    const int tile  = blockIdx.x * (blockDim.x >> 5) + waveId;
    const int tileM = tile / tilesN;
    const int tileN = tile - tileM * tilesN;

    const int lm   = lane & 15;
    const int half = lane >> 4;

    const float* __restrict__ arow = X + (tileM * 16 + lm) * K + 2 * half;
    const float* __restrict__ brow = W + (tileN * 16 + lm) * ldw + 2 * half;

    v8f acc = {};
#pragma unroll 4
    for (int k = 0; k < K; k += 4) {
        v2f a = *(const v2f*)(arow + k);
        v2f b = *(const v2f*)(brow + k);
        acc = __builtin_amdgcn_wmma_f32_16x16x4_f32(
            /*neg_a=*/false, a, /*neg_b=*/false, b,
            /*c_mod=*/(short)0, acc, /*reuse_a=*/false, /*reuse_b=*/false);
    }

    const int n  = tileN * 16 + lm;
    const float bn = bias ? bias[n] : 0.0f;
#pragma unroll
    for (int r = 0; r < 8; ++r) {
        const int m = tileM * 16 + r + 8 * half;
        float v = acc[r] + bn;
        if (relu) v = fmaxf(v, 0.0f);
        Y[m * Ncols + n] = v;
    }
}

// ---------------------------------------------------------------------------
// repl[i] = sigmoid( sum_k T[i][k] * w[k] + b[0] ), one wave32 per row.
// ---------------------------------------------------------------------------
__global__ void matvec_sigmoid(const float* __restrict__ T,
                               const float* __restrict__ w,
                               const float* __restrict__ b,
                               float* __restrict__ out, int K)
{
    const int lane = threadIdx.x & 31;
    const int row  = blockIdx.x * (blockDim.x >> 5) + (threadIdx.x >> 5);
    const float* __restrict__ trow = T + row * K;

    float s = 0.0f;
    for (int k = lane; k < K; k += 32) s = fmaf(trow[k], w[k], s);
#pragma unroll
    for (int off = 16; off > 0; off >>= 1) s += __shfl_xor(s, off, 32);

    if (lane == 0) {
        const float z = s + b[0];
        out[row] = 1.0f / (1.0f + expf(-z));
    }
}

// ---------------------------------------------------------------------------
// edge[i,j] = sum_k w[k] * max(hi[i,k] + hj[j,k], 0) + bu2
// (bu1 already folded into hi by the producing GEMM's bias.)
// Block: 256 threads -> 64x64 output tile, 4x4 register micro-tile/thread,
// LDS k-chunks of 64 (row stride 66: 8B-aligned, bank-conflict light).
// ---------------------------------------------------------------------------
#define ETI 64
#define EKC 64
#define EPAD 66

__global__ void edge_kernel(const float* __restrict__ hi,
                            const float* __restrict__ hj,
                            const float* __restrict__ w,
                            const float* __restrict__ bu2,
                            float* __restrict__ out, int Nn, int K)
{
    __shared__ float shi[ETI][EPAD];
    __shared__ float shj[ETI][EPAD];
    __shared__ float sw[EKC];

    const int tid = threadIdx.x;
    const int bi  = blockIdx.x * ETI;
    const int bj  = blockIdx.y * ETI;
    const int i0  = (tid >> 4) << 2;   // 0..60 step 4
    const int j0  = (tid & 15) << 2;   // 0..60 step 4

    float acc[4][4] = {};

    for (int kc = 0; kc < K; kc += EKC) {
        __syncthreads();
        // cooperative tile load: 64 rows x 32 float2 per matrix, 8 iters/thread
        for (int idx = tid; idx < ETI * (EKC / 2); idx += 256) {
            const int row = idx >> 5;
            const int c   = (idx & 31) << 1;
            *(float2*)&shi[row][c] = *(const float2*)(hi + (bi + row) * K + kc + c);
            *(float2*)&shj[row][c] = *(const float2*)(hj + (bj + row) * K + kc + c);
        }
        if (tid < EKC) sw[tid] = w[kc + tid];
        __syncthreads();

        for (int k = 0; k < EKC; k += 2) {
            float2 a[4], b[4];
#pragma unroll
            for (int t = 0; t < 4; ++t) {
                a[t] = *(const float2*)&shi[i0 + t][k];
                b[t] = *(const float2*)&shj[j0 + t][k];
            }
            const float2 wv = *(const float2*)&sw[k];
#pragma unroll
            for (int ti = 0; ti < 4; ++ti)
#pragma unroll
                for (int tj = 0; tj < 4; ++tj) {
                    acc[ti][tj] = fmaf(wv.x, fmaxf(a[ti].x + b[tj].x, 0.0f), acc[ti][tj]);
                    acc[ti][tj] = fmaf(wv.y, fmaxf(a[ti].y + b[tj].y, 0.0f), acc[ti][tj]);
                }
        }
    }

    const float eb = bu2[0];
#pragma unroll
    for (int ti = 0; ti < 4; ++ti) {
        float4 o;
        o.x = acc[ti][0] + eb;
        o.y = acc[ti][1] + eb;
        o.z = acc[ti][2] + eb;
        o.w = acc[ti][3] + eb;
        *(float4*)(out + (bi + i0 + ti) * Nn + bj + j0) = o;
    }
}

// ---------------------------------------------------------------------------
// Launch: 5 WMMA GEMMs + matvec + edge kernel, serialized on `stream`.
// ---------------------------------------------------------------------------
extern "C" void kernel_launch(void* const* d_in, const int* in_sizes, int n_in,
                              void* d_out, int out_size, void* d_ws, size_t ws_size,
                              hipStream_t stream)
{
    const float* X   = (const float*)d_in[0];   // node_embeddings (1024x256)
    // d_in[1] = adjacency_matrix (unused by reference)
    const float* W1  = (const float*)d_in[2];
    const float* b1  = (const float*)d_in[3];
    const float* W2  = (const float*)d_in[4];
    const float* b2  = (const float*)d_in[5];
    const float* Wr1 = (const float*)d_in[6];
    const float* br1 = (const float*)d_in[7];
    const float* Wr2 = (const float*)d_in[8];   // (1x256)
    const float* br2 = (const float*)d_in[9];   // (1,)
    const float* Wu1 = (const float*)d_in[10];  // (256 x 512)
    const float* bu1 = (const float*)d_in[11];  // (256,)
    const float* Wu2 = (const float*)d_in[12];  // (1x256)
    const float* bu2 = (const float*)d_in[13];  // (1,)

    const int UEL = GN * GD;                    // 262144
    float* ws = (float*)d_ws;
    float* t1 = ws;                             // 1 MB
    float* t2 = ws + UEL;                       // 1 MB
    float* hi = ws + 2 * UEL;                   // 1 MB
    float* hj = ws + 3 * UEL;                   // 1 MB

    float* u    = (float*)d_out;                // output 0
    float* repl = (float*)d_out + UEL;          // output 1
    float* edge = (float*)d_out + UEL + GN;     // output 2

    // GEMM launch shape: (1024/16)*(256/16)=1024 tiles, 4 waves/block -> 256 blocks
    const dim3 gB(128);
    const dim3 gG((GN / 16) * (GD / 16) / 4);

    // t1 = ReLU(X W1^T + b1)
    wmma_gemm_f32<<<gG, gB, 0, stream>>>(X,  W1, b1, t1, GN, GD, GD, GD, 1);
    // u = t1 W2^T + b2      (written directly into d_out)
    wmma_gemm_f32<<<gG, gB, 0, stream>>>(t1, W2, b2, u,  GN, GD, GD, GD, 0);
    // t2 = ReLU(u Wr1^T + br1)
    wmma_gemm_f32<<<gG, gB, 0, stream>>>(u, Wr1, br1, t2, GN, GD, GD, GD, 1);
    // repl = sigmoid(t2 . Wr2 + br2)   (8 waves/block -> 128 blocks)
    matvec_sigmoid<<<dim3(GN / 8), 256, 0, stream>>>(t2, Wr2, br2, repl, GD);
    // hi = u A^T + bu1  (A = Wu1[:, :D], row stride 512; bu1 folded here)
    wmma_gemm_f32<<<gG, gB, 0, stream>>>(u, Wu1,      bu1,     hi, GN, GD, GD, 2 * GD, 0);
    // hj = u B^T        (B = Wu1[:, D:], row stride 512)
    wmma_gemm_f32<<<gG, gB, 0, stream>>>(u, Wu1 + GD, nullptr, hj, GN, GD, GD, 2 * GD, 0);
    // edge[i,j] = sum_k w[k]*ReLU(hi+hj) + bu2
    edge_kernel<<<dim3(GN / ETI, GN / ETI), 256, 0, stream>>>(hi, hj, Wu2, bu2, edge, GN, GD);
}